// LinkTransformerLayer_7679401525464
// MI455X (gfx1250) — compile-verified
//
#include <hip/hip_runtime.h>
#include <hip/hip_bf16.h>

#define BSZ   8192
#define NV    50000
#define NE    131072
#define DD    128
#define NH    4
#define NC    128
#define HC    512           // NH*NC
#define NEG_SLOPE 0.2f
#define LN_EPS 1e-5f

typedef __attribute__((ext_vector_type(2))) float v2f;
typedef __attribute__((ext_vector_type(8))) float v8f;

// -------- ordered-float encoding for uint atomicMax --------
__device__ __forceinline__ unsigned ord_enc(float f) {
    unsigned u = __float_as_uint(f);
    return (u & 0x80000000u) ? ~u : (u | 0x80000000u);
}
__device__ __forceinline__ float ord_dec(unsigned u) {
    return (u & 0x80000000u) ? __uint_as_float(u & 0x7FFFFFFFu)
                             : __uint_as_float(~u);
}

__device__ __forceinline__ float wave_sum(float v) {
    #pragma unroll
    for (int m = 16; m; m >>= 1) v += __shfl_xor(v, m, 32);
    return v;
}

// -------- zero scratch (amax/denom/acc) --------
__global__ void zero_kernel(float* __restrict__ p, size_t n) {
    size_t i = (size_t)blockIdx.x * blockDim.x + threadIdx.x;
    size_t stride = (size_t)gridDim.x * blockDim.x;
    for (; i < n; i += stride) p[i] = 0.0f;
}

// -------- GEMM1: S = (e1+e2) @ W_l + 2*b_l   [8192 x 512], K=128 --------
// one wave -> 16 (M) x 64 (N) output strip; 8 waves per block cover N=512
__global__ __launch_bounds__(256)
void gemm_edge_kernel(const float* __restrict__ edge_x,
                      const float* __restrict__ W_l,
                      const float* __restrict__ b_l,
                      float* __restrict__ S) {
    const int lane = threadIdx.x & 31;
    const int wave = threadIdx.x >> 5;
    const int m0   = blockIdx.x * 16;       // 512 blocks
    const int n0   = wave * 64;
    const int mrow = lane & 15;
    const int hi   = lane >> 4;             // K-half selector
    const int ncol = lane & 15;

    v8f c[4] = {};
    const float* arow = edge_x + (size_t)(m0 + mrow) * (2 * DD);
    for (int kk = 0; kk < DD; kk += 4) {
        const int ka = kk + 2 * hi;
        v2f a;
        a.x = arow[ka]     + arow[DD + ka];
        a.y = arow[ka + 1] + arow[DD + ka + 1];
        #pragma unroll
        for (int t = 0; t < 4; ++t) {
            const float* bp = W_l + (size_t)ka * HC + n0 + t * 16 + ncol;
            v2f b; b.x = bp[0]; b.y = bp[HC];
            c[t] = __builtin_amdgcn_wmma_f32_16x16x4_f32(
                false, a, false, b, (short)0, c[t], false, false);
        }
    }
    #pragma unroll
    for (int t = 0; t < 4; ++t) {
        const int col = n0 + t * 16 + ncol;
        const float bb = 2.0f * b_l[col];
        #pragma unroll
        for (int r = 0; r < 8; ++r)
            S[(size_t)(m0 + r + 8 * hi) * HC + col] = c[t][r] + bb;
    }
}

// -------- GEMM2: XJ = node_x @ W_r + b_r   [50000 x 512], K=384 --------
__global__ __launch_bounds__(256)
void gemm_node_kernel(const float* __restrict__ node_x,
                      const float* __restrict__ W_r,
                      const float* __restrict__ b_r,
                      float* __restrict__ XJ) {
    const int lane = threadIdx.x & 31;
    const int wave = threadIdx.x >> 5;
    const int m0   = blockIdx.x * 16;       // 3125 blocks (50000/16)
    const int n0   = wave * 64;
    const int mrow = lane & 15;
    const int hi   = lane >> 4;
    const int ncol = lane & 15;

    v8f c[4] = {};
    const float* arow = node_x + (size_t)(m0 + mrow) * (3 * DD);
    for (int kk = 0; kk < 3 * DD; kk += 4) {
        const int ka = kk + 2 * hi;
        const v2f a = *(const v2f*)(arow + ka);      // contiguous K pair
        #pragma unroll
        for (int t = 0; t < 4; ++t) {
            const float* bp = W_r + (size_t)ka * HC + n0 + t * 16 + ncol;
            v2f b; b.x = bp[0]; b.y = bp[HC];
            c[t] = __builtin_amdgcn_wmma_f32_16x16x4_f32(
                false, a, false, b, (short)0, c[t], false, false);
        }
    }
    #pragma unroll
    for (int t = 0; t < 4; ++t) {
        const int col = n0 + t * 16 + ncol;
        const float bb = b_r[col];
        #pragma unroll
        for (int r = 0; r < 8; ++r)
            XJ[(size_t)(m0 + r + 8 * hi) * HC + col] = c[t][r] + bb;
    }
}

// -------- per-edge alpha + segment max (one wave per edge) --------
__global__ __launch_bounds__(256)
void edge_alpha_kernel(const int* __restrict__ eidx,
                       const float* __restrict__ XJ,
                       const float* __restrict__ S,
                       const float* __restrict__ att,
                       float* __restrict__ alpha,
                       unsigned* __restrict__ amax) {
    const int lane = threadIdx.x & 31;
    const int e = blockIdx.x * 8 + (threadIdx.x >> 5);
    const int seg = eidx[e];
    const int nid = eidx[NE + e];
    const float* xr = XJ + (size_t)nid * HC;
    const float* sr = S + (size_t)seg * HC;

    float acc[NH] = {0.f, 0.f, 0.f, 0.f};
    #pragma unroll
    for (int i = 0; i < 16; ++i) {
        const int j = i * 32 + lane;        // head = j>>7 = i>>2 (coalesced)
        float v = xr[j] * sr[j];
        v = (v > 0.f) ? v : NEG_SLOPE * v;  // leaky_relu
        acc[i >> 2] += v * att[j];          // att flat [H*C] indexed by j
    }
    #pragma unroll
    for (int h = 0; h < NH; ++h) acc[h] = wave_sum(acc[h]);
    if (lane < NH) {
        alpha[(size_t)e * NH + lane] = acc[lane];
        atomicMax(&amax[(size_t)seg * NH + lane], ord_enc(acc[lane]));
    }
}

// -------- exp + segment-sum denominator (one thread per (e,h)) --------
__global__ __launch_bounds__(256)
void edge_exp_kernel(const int* __restrict__ eidx,
                     float* __restrict__ alpha,
                     const unsigned* __restrict__ amax,
                     float* __restrict__ denom) {
    const int tid = blockIdx.x * blockDim.x + threadIdx.x;  // NE*NH threads
    const int e = tid >> 2, h = tid & 3;
    const int seg = eidx[e];
    const float m = ord_dec(amax[(size_t)seg * NH + h]);
    const float ex = __expf(alpha[tid] - m);
    alpha[tid] = ex;
    atomicAdd(&denom[(size_t)seg * NH + h], ex);
}

// -------- weighted scatter (one wave per edge, fp32 atomics into L2) ----
__global__ __launch_bounds__(256)
void edge_scatter_kernel(const int* __restrict__ eidx,
                         const float* __restrict__ XJ,
                         const float* __restrict__ alpha,
                         const float* __restrict__ denom,
                         float* __restrict__ acc) {
    const int lane = threadIdx.x & 31;
    const int e = blockIdx.x * 8 + (threadIdx.x >> 5);
    const int seg = eidx[e];
    const int nid = eidx[NE + e];
    const float* xr = XJ + (size_t)nid * HC;
    float w[NH];
    #pragma unroll
    for (int h = 0; h < NH; ++h)
        w[h] = alpha[(size_t)e * NH + h] / denom[(size_t)seg * NH + h];
    float* orow = acc + (size_t)seg * HC;
    #pragma unroll
    for (int i = 0; i < 16; ++i) {
        const int j = i * 32 + lane;
        atomicAdd(&orow[j], xr[j] * w[i >> 2]);
    }
}

// -------- bias + LayerNorm (one wave per row) --------
__global__ __launch_bounds__(256)
void layernorm_kernel(const float* __restrict__ acc,
                      const float* __restrict__ bias,
                      const float* __restrict__ gamma,
                      const float* __restrict__ beta,
                      float* __restrict__ out) {
    const int lane = threadIdx.x & 31;
    const int row = blockIdx.x * 8 + (threadIdx.x >> 5);
    const float* r = acc + (size_t)row * HC;
    float v[16], s = 0.f, s2 = 0.f;
    #pragma unroll
    for (int i = 0; i < 16; ++i) {
        const int j = i * 32 + lane;
        v[i] = r[j] + bias[j];
        s += v[i]; s2 += v[i] * v[i];
    }
    s = wave_sum(s); s2 = wave_sum(s2);
    const float mu = s * (1.0f / HC);
    const float var = s2 * (1.0f / HC) - mu * mu;
    const float rn = rsqrtf(var + LN_EPS);
    float* o = out + (size_t)row * HC;
    #pragma unroll
    for (int i = 0; i < 16; ++i) {
        const int j = i * 32 + lane;
        o[j] = (v[i] - mu) * rn * gamma[j] + beta[j];
    }
}

extern "C" void kernel_launch(void* const* d_in, const int* in_sizes, int n_in,
                              void* d_out, int out_size, void* d_ws, size_t ws_size,
                              hipStream_t stream) {
    const float* edge_x = (const float*)d_in[0];
    const float* node_x = (const float*)d_in[1];
    const float* W_l    = (const float*)d_in[2];
    const float* b_l    = (const float*)d_in[3];
    const float* W_r    = (const float*)d_in[4];
    const float* b_r    = (const float*)d_in[5];
    const float* att    = (const float*)d_in[6];
    const float* bias   = (const float*)d_in[7];
    const float* gamma  = (const float*)d_in[8];
    const float* beta   = (const float*)d_in[9];
    const int*   eidx   = (const int*)d_in[10];

    // workspace layout (floats)
    float* ws     = (float*)d_ws;
    float* s_buf  = ws;                                    // BSZ*HC
    float* xj     = s_buf + (size_t)BSZ * HC;              // NV*HC
    float* alpha  = xj    + (size_t)NV * HC;               // NE*NH
    unsigned* amax = (unsigned*)(alpha + (size_t)NE * NH); // BSZ*NH
    float* denom  = (float*)(amax + (size_t)BSZ * NH);     // BSZ*NH
    float* acc    = denom + (size_t)BSZ * NH;              // BSZ*HC

    // zero amax+denom+acc (contiguous)
    const size_t nz = (size_t)BSZ * NH * 2 + (size_t)BSZ * HC;
    zero_kernel<<<2048, 256, 0, stream>>>((float*)amax, nz);

    gemm_edge_kernel<<<BSZ / 16, 256, 0, stream>>>(edge_x, W_l, b_l, s_buf);
    gemm_node_kernel<<<NV / 16, 256, 0, stream>>>(node_x, W_r, b_r, xj);
    edge_alpha_kernel<<<NE / 8, 256, 0, stream>>>(eidx, xj, s_buf, att, alpha, amax);
    edge_exp_kernel<<<(NE * NH) / 256, 256, 0, stream>>>(eidx, alpha, amax, denom);
    edge_scatter_kernel<<<NE / 8, 256, 0, stream>>>(eidx, xj, alpha, denom, acc);
    layernorm_kernel<<<BSZ / 8, 256, 0, stream>>>(acc, bias, gamma, beta, (float*)d_out);
}